// Waveletatt_afterlayer_75307956568684
// MI455X (gfx1250) — compile-verified
//
#include <hip/hip_runtime.h>
#include <hip/hip_bf16.h>
#include <math.h>

typedef __attribute__((ext_vector_type(2))) float v2f;
typedef __attribute__((ext_vector_type(4))) float v4f;
typedef __attribute__((ext_vector_type(8))) float v8f;

#define BDIM   16
#define CDIM   256
#define HDIM   224
#define WDIM   224
#define PLANE  (HDIM * WDIM)   // 50176 floats per (b,c) plane
#define PLANE4 (PLANE / 4)     // 12544 float4 per plane
#define W4     (WDIM / 4)      // 56 float4 per row

// ---------------------------------------------------------------------------
// Kernel 1: per-(b,c) plane reduction -> 4 wavelet band means (ori_y).
// One block per plane; nontemporal float4 streaming; parity-split sums.
// ---------------------------------------------------------------------------
__global__ __launch_bounds__(256)
void wr_reduce(const float* __restrict__ x, float* __restrict__ oriY) {
  const int plane = blockIdx.x;                       // b*256 + c
  const v4f* xp = (const v4f*)(x + (size_t)plane * PLANE);

  float s_ee = 0.f, s_oe = 0.f, s_eo = 0.f, s_oo = 0.f;
  for (int i = threadIdx.x; i < PLANE4; i += 256) {
    const int row = i / W4;
    v4f v = __builtin_nontemporal_load(xp + i);
    const float ev = v.x + v.z;   // even columns within this float4
    const float od = v.y + v.w;   // odd columns
    if (row & 1) { s_oe += ev; s_oo += od; }
    else         { s_ee += ev; s_eo += od; }
  }

  __shared__ v4f red[256];
  v4f mine = {s_ee, s_oe, s_eo, s_oo};
  red[threadIdx.x] = mine;
  __syncthreads();
  for (int off = 128; off > 0; off >>= 1) {
    if ((int)threadIdx.x < off) red[threadIdx.x] += red[threadIdx.x + off];
    __syncthreads();
  }

  if (threadIdx.x == 0) {
    const float inv = 0.5f / 12544.f;     // *0.5 scale, /(112*112) mean
    const float m1 = red[0].x * inv;      // even row, even col  (x1)
    const float m2 = red[0].y * inv;      // odd  row, even col  (x2)
    const float m3 = red[0].z * inv;      // even row, odd  col  (x3)
    const float m4 = red[0].w * inv;      // odd  row, odd  col  (x4)
    const int b = plane >> 8, c = plane & 255;
    float* o = oriY + ((size_t)b * 4) * CDIM + c;
    o[0 * CDIM] =  m1 + m2 + m3 + m4;     // LL
    o[1 * CDIM] = -m1 - m2 + m3 + m4;     // HL
    o[2 * CDIM] = -m1 + m2 - m3 + m4;     // LH
    o[3 * CDIM] =  m1 - m2 - m3 + m4;     // HH
  }
}

// ---------------------------------------------------------------------------
// Kernel 2: gate computation on one 4-wave block.
//   logits -> argmax -> y = sum_k relu(ori_y - Q)
//   h = relu(y @ w1^T)  via V_WMMA_F32_16X16X4_F32   (M=16, K=256, N=128)
//   s = sigmoid(h @ w2^T) via WMMA                   (M=16, K=128, N=256)
// ---------------------------------------------------------------------------
__global__ __launch_bounds__(128)
void wr_gate(const float* __restrict__ oriY,
             const float* __restrict__ w1,     // (128, 256) row-major
             const float* __restrict__ w2,     // (256, 128) row-major
             const float* __restrict__ wenc,   // (256)
             const float* __restrict__ benc,   // (1)
             float* __restrict__ s_out) {      // (16, 256)
  __shared__ float sY[BDIM * 4 * CDIM];   // 64 KB  ori_y
  __shared__ float sy[BDIM * CDIM];       // 16 KB  y  (16 x 256)
  __shared__ float sh[BDIM * 128];        //  8 KB  h  (16 x 128)
  __shared__ float slog[BDIM * 4];
  __shared__ int   skmax[BDIM];
  const int tid = threadIdx.x;

  for (int i = tid; i < BDIM * 4 * CDIM; i += 128) sY[i] = oriY[i];
  __syncthreads();

  // 64 logits, one per thread
  if (tid < 64) {
    const int b = tid >> 2, k = tid & 3;
    const float* row = &sY[(b * 4 + k) * CDIM];
    float acc = 0.f;
    for (int c = 0; c < CDIM; ++c) acc += row[c] * wenc[c];
    slog[tid] = acc + benc[0];
  }
  __syncthreads();

  // argmax over k (sigmoid is monotonic; strict > keeps first max like jnp)
  if (tid < BDIM) {
    float best = slog[tid * 4]; int kb = 0;
    for (int k = 1; k < 4; ++k) {
      const float v = slog[tid * 4 + k];
      if (v > best) { best = v; kb = k; }
    }
    skmax[tid] = kb;
  }
  __syncthreads();

  // y[b][c] = sum_k relu(ori_y[b,k,c] - Q[b,c])
  for (int i = tid; i < BDIM * CDIM; i += 128) {
    const int b = i >> 8, c = i & 255;
    const float q = sY[(b * 4 + skmax[b]) * CDIM + c];
    float acc = 0.f;
    for (int k = 0; k < 4; ++k) acc += fmaxf(sY[(b * 4 + k) * CDIM + c] - q, 0.f);
    sy[i] = acc;
  }
  __syncthreads();

  const int wave = tid >> 5;
  const int lane = tid & 31;
  const int m  = lane & 15;           // A-matrix row (batch index)
  const int hi = lane >> 4;           // lane half selects K pair
  const int kg = hi << 1;             // K sub-offset {0, 2}

  // GEMM1: h = relu(y @ w1^T); 8 column tiles, 2 per wave
  for (int t = 0; t < 2; ++t) {
    const int n = (wave * 2 + t) * 16 + m;
    v8f acc = {};
    for (int k0 = 0; k0 < 256; k0 += 4) {
      v2f a, bfrag;
      a.x     = sy[m * 256 + k0 + kg];
      a.y     = sy[m * 256 + k0 + kg + 1];
      bfrag.x = w1[n * 256 + k0 + kg];       // B[k][n] = w1[n][k]
      bfrag.y = w1[n * 256 + k0 + kg + 1];
      acc = __builtin_amdgcn_wmma_f32_16x16x4_f32(
          false, a, false, bfrag, (short)0, acc, false, false);
    }
#pragma unroll
    for (int r = 0; r < 8; ++r) {
      const int mm = r + 8 * hi;             // D layout: M = r + 8*(lane/16)
      sh[mm * 128 + n] = fmaxf(acc[r], 0.f);
    }
  }
  __syncthreads();

  // GEMM2: s = sigmoid(h @ w2^T); 16 column tiles, 4 per wave
  for (int t = 0; t < 4; ++t) {
    const int n = (wave * 4 + t) * 16 + m;
    v8f acc = {};
    for (int k0 = 0; k0 < 128; k0 += 4) {
      v2f a, bfrag;
      a.x     = sh[m * 128 + k0 + kg];
      a.y     = sh[m * 128 + k0 + kg + 1];
      bfrag.x = w2[n * 128 + k0 + kg];       // B[k][n] = w2[n][k]
      bfrag.y = w2[n * 128 + k0 + kg + 1];
      acc = __builtin_amdgcn_wmma_f32_16x16x4_f32(
          false, a, false, bfrag, (short)0, acc, false, false);
    }
#pragma unroll
    for (int r = 0; r < 8; ++r) {
      const int mm = r + 8 * hi;
      const float v = acc[r];
      s_out[mm * 256 + n] = 1.f / (1.f + __expf(-v));
    }
  }
}

// ---------------------------------------------------------------------------
// Kernel 3: out = x * s[b,c]; one block per plane, NT float4 streaming.
// ---------------------------------------------------------------------------
__global__ __launch_bounds__(256)
void wr_scale(const float* __restrict__ x, const float* __restrict__ s,
              float* __restrict__ out) {
  const int plane = blockIdx.x;
  const float sc = s[plane];
  const v4f* xp = (const v4f*)(x + (size_t)plane * PLANE);
  v4f* op = (v4f*)(out + (size_t)plane * PLANE);
  for (int i = threadIdx.x; i < PLANE4; i += 256) {
    v4f v = __builtin_nontemporal_load(xp + i);
    v = v * sc;
    __builtin_nontemporal_store(v, op + i);
  }
}

// ---------------------------------------------------------------------------
extern "C" void kernel_launch(void* const* d_in, const int* in_sizes, int n_in,
                              void* d_out, int out_size, void* d_ws, size_t ws_size,
                              hipStream_t stream) {
  const float* x    = (const float*)d_in[0];
  const float* w1   = (const float*)d_in[1];
  const float* w2   = (const float*)d_in[2];
  const float* wenc = (const float*)d_in[3];
  const float* benc = (const float*)d_in[4];
  float* out = (float*)d_out;

  float* oriY  = (float*)d_ws;                 // 16*4*256 floats = 64 KB
  float* sgate = oriY + BDIM * 4 * CDIM;       // 16*256 floats   = 16 KB

  wr_reduce<<<BDIM * CDIM, 256, 0, stream>>>(x, oriY);
  wr_gate<<<1, 128, 0, stream>>>(oriY, w1, w2, wenc, benc, sgate);
  wr_scale<<<BDIM * CDIM, 256, 0, stream>>>(x, sgate, out);
}